// FlexibleAttentionModel_75892072121057
// MI455X (gfx1250) — compile-verified
//
#include <hip/hip_runtime.h>
#include <math.h>

#define B_    4
#define S_    512
#define V_    32000
#define D_    256
#define H_    4
#define DH_   64
#define NEUR_ 128
#define LUTN_ 4096

static constexpr float PHI_F     = 1.618033988749895f;
static constexpr float TWO_PI_F  = 6.283185307179586f;
static constexpr float LUT_SCALE = (float)LUTN_ / TWO_PI_F;
static constexpr float INV_SCALE = 0.08838834764831845f; // 1/sqrt(2*DH)=1/sqrt(128)

typedef float v2f __attribute__((ext_vector_type(2)));
typedef float v8f __attribute__((ext_vector_type(8)));

// ---------------------------------------------------------------------------
// Quantized LUT sin/cos, matching the reference (round-half-even, mod 4096)
// ---------------------------------------------------------------------------
__device__ __forceinline__ void lut_sc(float theta,
                                       const float* __restrict__ lutS,
                                       const float* __restrict__ lutC,
                                       float& s, float& c) {
  int idx = (int)rintf(theta * LUT_SCALE);
  idx &= (LUTN_ - 1);           // power-of-two mod works for negatives too
  s = lutS[idx];
  c = lutC[idx];
}

__global__ void lut_init_kernel(float* __restrict__ lutS, float* __restrict__ lutC) {
  int i = blockIdx.x * blockDim.x + threadIdx.x;
  if (i < LUTN_) {
    double th = 6.283185307179586476925286766559 * (double)i / (double)LUTN_;
    lutS[i] = (float)sin(th);
    lutC[i] = (float)cos(th);
  }
}

// ---------------------------------------------------------------------------
// Sequential scan: one workgroup, 1024 threads = one thread per (b,d).
// Produces X[t][b][d] (final per-step state) and K-cache (keys computed ONCE
// per position instead of O(S^2) as in the reference).
// ---------------------------------------------------------------------------
__global__ __launch_bounds__(1024) void seq_kernel(
    const int*   __restrict__ input_ids,   // (B,S)
    const float* __restrict__ embedding,   // (V,2D)
    const float* __restrict__ w_query,     // (H,DH)
    const float* __restrict__ b_query,     // (H,DH)
    const float* __restrict__ w_key,       // (H,DH)
    const float* __restrict__ b_key,       // (H,DH)
    const float* __restrict__ context_w,   // (D,D)
    const float* __restrict__ W_res,       // (NEUR,D)
    const float* __restrict__ B_res,       // (NEUR,D)
    const float* __restrict__ proj_real,   // (D,NEUR)
    const float* __restrict__ proj_imag,   // (D,NEUR)
    const float* __restrict__ lutS,
    const float* __restrict__ lutC,
    float* __restrict__ X,                 // (S,B,D)  workspace
    float* __restrict__ Kc)                // (B,S,H,2*DH) workspace
{
  __shared__ float sh_q  [B_ * H_ * 2 * DH_];   //  8 KB
  __shared__ float sh_sc [B_ * H_ * S_];        // 32 KB  scores -> softmax wts
  __shared__ float sh_ws [B_ * S_];             //  8 KB  sum over heads
  __shared__ float sh_x  [B_ * D_];             //  4 KB  post-context x
  __shared__ float sh_ctx[B_ * D_];             //  4 KB
  __shared__ float sh_cs [B_ * NEUR_];          //  2 KB
  __shared__ float sh_ss [B_ * NEUR_];          //  2 KB

  const int tid  = threadIdx.x;
  const int b    = tid >> 8;       // batch 0..3
  const int d    = tid & 255;      // feature 0..255
  const int h    = d >> 6;         // head 0..3
  const int dh   = d & 63;
  const int lane = tid & 31;
  const int warp = tid >> 5;       // 0..31

  // per-thread time-invariant params
  const float wlq = 1.0f + fabsf(w_query[h * DH_ + dh]);
  const float bq  = b_query[h * DH_ + dh];
  const float wlk = 1.0f + fabsf(w_key[h * DH_ + dh]);
  const float bk  = b_key[h * DH_ + dh];

  float h_real = 0.0f, h_imag = 0.0f;

  for (int t = 0; t < S_; ++t) {
    const float ts = (float)t * PHI_F;

    // ---- Phase A: recurrence + query ------------------------------------
    const int tok = input_ids[b * S_ + t];
    const float w_t = embedding[(size_t)tok * (2 * D_) + d];
    const float b_t = embedding[(size_t)tok * (2 * D_) + D_ + d];
    const float wl  = 1.0f + fabsf(w_t);
    float sr, cr, si, ci;
    lut_sc(h_real / wl + b_t + ts, lutS, lutC, sr, cr);
    lut_sc(h_imag / wl + b_t + ts, lutS, lutC, si, ci);
    h_real = cr * ci - sr * si;
    h_imag = cr * si + sr * ci;
    const float x_pre = h_real + h_imag;

    float sq, cq;
    lut_sc(x_pre / wlq + bq + ts, lutS, lutC, sq, cq);
    sh_q[(b * H_ + h) * (2 * DH_) + dh]       = cq;
    sh_q[(b * H_ + h) * (2 * DH_) + DH_ + dh] = sq;
    __syncthreads();

    // ---- Phase B: attention (only s < t contribute) ---------------------
    float ctx = 0.0f;
    if (t > 0) {
      // scores: 2 warps per (b,h); lanes stride over s
      {
        const int p = warp >> 1, half = warp & 1;
        const int bb = p >> 2, hh = p & 3;
        const float4* q4 = (const float4*)&sh_q[(bb * H_ + hh) * (2 * DH_)];
        for (int s = half * 32 + lane; s < t; s += 64) {
          const float4* k4 =
              (const float4*)&Kc[(((size_t)bb * S_ + s) * H_ + hh) * (2 * DH_)];
          float acc = 0.0f;
#pragma unroll 8
          for (int j = 0; j < 32; ++j) {
            float4 qv = q4[j];
            float4 kv = k4[j];
            acc += qv.x * kv.x + qv.y * kv.y + qv.z * kv.z + qv.w * kv.w;
          }
          sh_sc[(bb * H_ + hh) * S_ + s] = acc * INV_SCALE;
        }
      }
      __syncthreads();

      // softmax per (b,h): one warp each (warps 0..15)
      if (warp < 16) {
        const int bb = warp >> 2, hh = warp & 3;
        float* sc = &sh_sc[(bb * H_ + hh) * S_];
        float m = -1e30f;
        for (int s = lane; s < t; s += 32) m = fmaxf(m, sc[s]);
#pragma unroll
        for (int off = 16; off; off >>= 1) m = fmaxf(m, __shfl_xor(m, off, 32));
        float sum = 0.0f;
        for (int s = lane; s < t; s += 32) {
          float e = __expf(sc[s] - m);
          sc[s] = e;
          sum += e;
        }
#pragma unroll
        for (int off = 16; off; off >>= 1) sum += __shfl_xor(sum, off, 32);
        const float inv = 1.0f / sum;
        for (int s = lane; s < t; s += 32) sc[s] *= inv;
      }
      __syncthreads();

      // wsum[b][s] = sum over heads of weights
      for (int i = tid; i < B_ * S_; i += 1024) {
        const int bb = i >> 9, s = i & 511;
        if (s < t) {
          float v = 0.0f;
#pragma unroll
          for (int hh = 0; hh < H_; ++hh) v += sh_sc[(bb * H_ + hh) * S_ + s];
          sh_ws[i] = v;
        }
      }
      __syncthreads();

      // context[b][d] = sum_s wsum[b][s] * X[s][b][d]   (coalesced over d)
      for (int s = 0; s < t; ++s)
        ctx += sh_ws[b * S_ + s] * X[((size_t)s * B_ + b) * D_ + d];
      sh_ctx[b * D_ + d] = ctx;
      __syncthreads();
    }

    // ---- Phase C: x += context @ context_w^T ----------------------------
    float xn = x_pre;
    if (t > 0) {
      const float* cw   = &context_w[(size_t)d * D_];   // row d
      const float* ctxv = &sh_ctx[b * D_];
      float a = 0.0f;
#pragma unroll 4
      for (int j = 0; j < D_; ++j) a += ctxv[j] * cw[j];
      xn = x_pre + a;
    }
    sh_x[b * D_ + d] = xn;
    __syncthreads();

    // ---- Phase D: reservoir sums csum/ssum (warp per 16 (b,n) pairs) ----
#pragma unroll 1
    for (int i = 0; i < 16; ++i) {
      const int pi = warp * 16 + i;          // pi = b*128 + n
      const int bb = pi >> 7, n = pi & 127;
      float cs = 0.0f, ss = 0.0f;
      for (int dd = lane; dd < D_; dd += 32) {   // coalesced over lanes
        const float wr = 1.0f + fabsf(W_res[n * D_ + dd]);
        const float br = B_res[n * D_ + dd];
        float sv, cv;
        lut_sc(sh_x[bb * D_ + dd] / wr + br + ts, lutS, lutC, sv, cv);
        cs += cv;
        ss += sv;
      }
#pragma unroll
      for (int off = 16; off; off >>= 1) {
        cs += __shfl_xor(cs, off, 32);
        ss += __shfl_xor(ss, off, 32);
      }
      if (lane == 0) { sh_cs[pi] = cs; sh_ss[pi] = ss; }
    }
    __syncthreads();

    // ---- Phase E: SiLU residual, commit state + key cache ---------------
    float y = 0.0f;
    {
      const float* pr  = &proj_real[(size_t)d * NEUR_];
      const float* pim = &proj_imag[(size_t)d * NEUR_];
      const float* csv = &sh_cs[b * NEUR_];
      const float* ssv = &sh_ss[b * NEUR_];
#pragma unroll 4
      for (int n = 0; n < NEUR_; ++n) y += csv[n] * pr[n] + ssv[n] * pim[n];
    }
    const float sig = 1.0f / (1.0f + __expf(-y));
    const float xf  = xn + y * sig;

    X[((size_t)t * B_ + b) * D_ + d] = xf;

    // key for position t computed once (time-invariant, reused by all t'>t)
    float sk, ck;
    lut_sc(xf / wlk + bk, lutS, lutC, sk, ck);
    Kc[(((size_t)b * S_ + t) * H_ + h) * (2 * DH_) + dh]       = ck;
    Kc[(((size_t)b * S_ + t) * H_ + h) * (2 * DH_) + DH_ + dh] = sk;
    __syncthreads();   // global writes visible block-wide for next step
  }
}

// ---------------------------------------------------------------------------
// Deferred logits GEMM: out[b][s][v] = sum_d X[s][b][d] * Wout[v][d]
// One wave per 64x64 tile: 4x4 grid of 16x16 f32 WMMA fragments,
// k-loop of 64 steps of V_WMMA_F32_16X16X4_F32 (f32-exact, L2-BW bound).
// ---------------------------------------------------------------------------
__global__ __launch_bounds__(128) void gemm_kernel(
    const float* __restrict__ X,      // (S,B,D)
    const float* __restrict__ Wout,   // (V,D)
    float* __restrict__ out)          // (B,S,V)
{
  const int lane  = threadIdx.x & 31;
  const int wv    = threadIdx.x >> 5;             // wave in block: 0..3
  const int rBase = blockIdx.y * 64;              // 32 row groups
  const int vBase = blockIdx.x * 256 + wv * 64;   // 125 col blocks x 4 waves

  const int mrow  = lane & 15;
  const int khalf = (lane >> 4) << 1;             // K sub-offset 0 or 2

  // A: rows are flattened (b,s) with row = b*S + s; X stored [s][b][d]
  const float* arow[4];
#pragma unroll
  for (int mi = 0; mi < 4; ++mi) {
    const int r  = rBase + mi * 16 + mrow;
    const int bb = r >> 9;          // / S_
    const int s  = r & (S_ - 1);
    arow[mi] = &X[((size_t)s * B_ + bb) * D_];
  }
  const float* brow[4];
#pragma unroll
  for (int ni = 0; ni < 4; ++ni) {
    const int v = vBase + ni * 16 + (lane & 15);
    brow[ni] = &Wout[(size_t)v * D_];
  }

  v8f acc[4][4];
#pragma unroll
  for (int mi = 0; mi < 4; ++mi)
#pragma unroll
    for (int ni = 0; ni < 4; ++ni) acc[mi][ni] = v8f{};

#pragma unroll 2
  for (int k = 0; k < D_; k += 4) {
    v2f a[4], bf[4];
#pragma unroll
    for (int mi = 0; mi < 4; ++mi) a[mi] = *(const v2f*)&arow[mi][k + khalf];
#pragma unroll
    for (int ni = 0; ni < 4; ++ni) bf[ni] = *(const v2f*)&brow[ni][k + khalf];
#pragma unroll
    for (int mi = 0; mi < 4; ++mi)
#pragma unroll
      for (int ni = 0; ni < 4; ++ni)
        acc[mi][ni] = __builtin_amdgcn_wmma_f32_16x16x4_f32(
            /*neg_a=*/false, a[mi], /*neg_b=*/false, bf[ni],
            /*c_mod=*/(short)0, acc[mi][ni],
            /*reuse_a=*/false, /*reuse_b=*/false);
  }

  // Store: C/D layout -> lane L: N = L&15, VGPR r: M = r + 8*(L>=16)
  const int ncol = lane & 15;
  const int moff = (lane >> 4) << 3;
#pragma unroll
  for (int mi = 0; mi < 4; ++mi)
#pragma unroll
    for (int ni = 0; ni < 4; ++ni) {
      const int vcol = vBase + ni * 16 + ncol;
#pragma unroll
      for (int r = 0; r < 8; ++r) {
        const int row = rBase + mi * 16 + moff + r;
        const int bb  = row >> 9;
        const int s   = row & (S_ - 1);
        out[((size_t)bb * S_ + s) * V_ + vcol] = acc[mi][ni][r];
      }
    }
}

// ---------------------------------------------------------------------------
extern "C" void kernel_launch(void* const* d_in, const int* in_sizes, int n_in,
                              void* d_out, int out_size, void* d_ws, size_t ws_size,
                              hipStream_t stream) {
  const int*   input_ids = (const int*)  d_in[0];
  const float* embedding = (const float*)d_in[1];
  const float* w_query   = (const float*)d_in[2];
  const float* b_query   = (const float*)d_in[3];
  const float* w_key     = (const float*)d_in[4];
  const float* b_key     = (const float*)d_in[5];
  const float* context_w = (const float*)d_in[6];
  const float* W_res     = (const float*)d_in[7];
  const float* B_res     = (const float*)d_in[8];
  const float* proj_real = (const float*)d_in[9];
  const float* proj_imag = (const float*)d_in[10];
  const float* output_w  = (const float*)d_in[11];

  float* ws   = (float*)d_ws;
  float* lutS = ws;                       // 4096 f32
  float* lutC = lutS + LUTN_;             // 4096 f32
  float* X    = lutC + LUTN_;             // S*B*D   = 524288 f32 (2 MB)
  float* Kc   = X + (size_t)S_ * B_ * D_; // B*S*H*128 = 1048576 f32 (4 MB)

  lut_init_kernel<<<LUTN_ / 256, 256, 0, stream>>>(lutS, lutC);

  seq_kernel<<<1, 1024, 0, stream>>>(input_ids, embedding, w_query, b_query,
                                     w_key, b_key, context_w, W_res, B_res,
                                     proj_real, proj_imag, lutS, lutC, X, Kc);

  gemm_kernel<<<dim3(V_ / 256, (B_ * S_) / 64), 128, 0, stream>>>(
      X, output_w, (float*)d_out);
}